// GNN_COO_22127671509527
// MI455X (gfx1250) — compile-verified
//
#include <hip/hip_runtime.h>
#include <hip/hip_bf16.h>

typedef __attribute__((ext_vector_type(2))) float v2f;
typedef __attribute__((ext_vector_type(8))) float v8f;

#define GCN_D 64          // feature dim (both layers)
#define GCN_NCLASS 10

// ---------------------------------------------------------------------------
// Degree / normalization kernels
// ---------------------------------------------------------------------------
__global__ void gcn_init_deg(float* __restrict__ deg, int n) {
    int i = blockIdx.x * blockDim.x + threadIdx.x;
    if (i < n) deg[i] = 1.0f;   // self-loop contributes 1 to every node's degree
}

__global__ void gcn_deg_accum(const int* __restrict__ dst, float* __restrict__ deg, int e) {
    int i = blockIdx.x * blockDim.x + threadIdx.x;
    if (i < e) atomicAdd(&deg[dst[i]], 1.0f);
}

__global__ void gcn_deg_to_dinv(float* __restrict__ deg, int n) {
    int i = blockIdx.x * blockDim.x + threadIdx.x;
    if (i < n) {
        float d = deg[i];
        deg[i] = (d > 0.0f) ? rsqrtf(d) : 0.0f;
    }
}

__global__ void gcn_zero(float* __restrict__ p, int n) {
    int i = blockIdx.x * blockDim.x + threadIdx.x;
    if (i < n) p[i] = 0.0f;
}

// ---------------------------------------------------------------------------
// WMMA fp32 GEMM:  out[m][0..63] = (X[m][:] @ W) * dinv[m]
// X: [nrows x 64] row-major, W: [64 x 64] row-major (K-major).
// Block = 64 threads = 2 waves; each wave produces a 16x64 tile via
// V_WMMA_F32_16X16X4_F32 (16 K-steps of K=4, 4 N-tiles).
//
// W is staged into LDS *K-pair interleaved*:
//   wlds[(k>>1)*128 + n*2 + (k&1)] = W[k][n]
// so that the B fragment {W[kk][n], W[kk+1][n]} (kk = k + 2*lhalf) is one
// contiguous, 8-byte-aligned ds_load_b64 per lane — no VGPR shuffle movs.
// ---------------------------------------------------------------------------
__global__ __launch_bounds__(64)
void gcn_gemm64_scale(const float* __restrict__ X, const float* __restrict__ W,
                      const float* __restrict__ dinv, float* __restrict__ out,
                      int nrows) {
    __shared__ float wlds[GCN_D * GCN_D];   // 16 KB, interleaved layout
    int tid = threadIdx.x;
    for (int i = tid; i < GCN_D * GCN_D; i += 64) {
        int k = i >> 6;          // row of W (K index)
        int n = i & 63;          // col of W (N index)
        wlds[(k >> 1) * 128 + n * 2 + (k & 1)] = W[i];
    }
    __syncthreads();

    int wave  = tid >> 5;
    int lane  = tid & 31;
    int lhalf = lane >> 4;     // 0: lanes 0-15, 1: lanes 16-31
    int l15   = lane & 15;
    int row0  = blockIdx.x * 32 + wave * 16;
    if (row0 >= nrows) return;                       // wave-uniform exit

    // clamp A-row for tail blocks (keeps EXEC all-ones through the WMMA loop)
    int arow = row0 + l15;
    if (arow >= nrows) arow = nrows - 1;
    const float* xrow = X + (size_t)arow * GCN_D;

    v8f c0 = {}, c1 = {}, c2 = {}, c3 = {};

    // A 16x4 fp32 layout: lanes 0-15 hold (M=l15, K=k..k+1), lanes 16-31 hold K=k+2..k+3
    // B 4x16 fp32 layout: lanes 0-15 hold rows {k,k+1}, lanes 16-31 rows {k+2,k+3}, col n0+l15
    for (int k = 0; k < GCN_D; k += 4) {
        v2f a = *(const v2f*)(xrow + k + lhalf * 2);
        // interleaved LDS base for K-pair (k/2 + lhalf), column l15 (+ n0)
        const v2f* wp = (const v2f*)(wlds + ((k >> 1) + lhalf) * 128 + l15 * 2);
        v2f b0 = wp[0];
        v2f b1 = wp[16];
        v2f b2 = wp[32];
        v2f b3 = wp[48];
        c0 = __builtin_amdgcn_wmma_f32_16x16x4_f32(false, a, false, b0, (short)0, c0, false, false);
        c1 = __builtin_amdgcn_wmma_f32_16x16x4_f32(false, a, false, b1, (short)0, c1, false, false);
        c2 = __builtin_amdgcn_wmma_f32_16x16x4_f32(false, a, false, b2, (short)0, c2, false, false);
        c3 = __builtin_amdgcn_wmma_f32_16x16x4_f32(false, a, false, b3, (short)0, c3, false, false);
    }

    // C/D 16x16 f32 layout: VGPR r -> (M = r + 8*lhalf, N = l15)
    #pragma unroll
    for (int r = 0; r < 8; ++r) {
        int m = row0 + r + lhalf * 8;
        if (m < nrows) {
            float s = dinv ? dinv[m] : 1.0f;
            float* orow = out + (size_t)m * GCN_D + l15;
            orow[0]  = c0[r] * s;
            orow[16] = c1[r] * s;
            orow[32] = c2[r] * s;
            orow[48] = c3[r] * s;
        }
    }
}

// ---------------------------------------------------------------------------
// Edge scatter: agg[dst][j] += Hs[src][j] * dinv[dst]   (Hs already has dinv[src])
// 64 lanes per edge -> coalesced gather + contiguous atomics.
// ---------------------------------------------------------------------------
__global__ __launch_bounds__(256)
void gcn_scatter(const int* __restrict__ src, const int* __restrict__ dst,
                 const float* __restrict__ hs, const float* __restrict__ dinv,
                 float* __restrict__ agg, int e) {
    int gid  = blockIdx.x * 256 + threadIdx.x;
    int eidx = gid >> 6;
    int j    = gid & 63;
    if (eidx >= e) return;
    int s = src[eidx];
    int d = dst[eidx];
    float v = hs[(size_t)s * GCN_D + j] * dinv[d];
    atomicAdd(&agg[(size_t)d * GCN_D + j], v);
}

// ---------------------------------------------------------------------------
// Epilogue: X'[i][j] = relu(agg[i][j] + Hs[i][j]*dinv[i] + b[j])   (self-loop fused)
// In-place over agg is safe (pure elementwise).
// ---------------------------------------------------------------------------
__global__ __launch_bounds__(256)
void gcn_finish_conv(const float* __restrict__ agg, const float* __restrict__ hs,
                     const float* __restrict__ dinv, const float* __restrict__ b,
                     float* __restrict__ out, int n) {
    int gid = blockIdx.x * 256 + threadIdx.x;
    int i = gid >> 6;
    int j = gid & 63;
    if (i >= n) return;
    float v = agg[gid] + hs[gid] * dinv[i] + b[j];
    out[gid] = fmaxf(v, 0.0f);
}

// ---------------------------------------------------------------------------
// Head: logits = X @ Wl + bl (64 -> 10), then log_softmax per node.
// ---------------------------------------------------------------------------
__global__ __launch_bounds__(256)
void gcn_head(const float* __restrict__ X, const float* __restrict__ Wl,
              const float* __restrict__ bl, float* __restrict__ out, int n) {
    __shared__ float wl[GCN_D * GCN_NCLASS];
    __shared__ float bs[GCN_NCLASS];
    for (int i = threadIdx.x; i < GCN_D * GCN_NCLASS; i += 256) wl[i] = Wl[i];
    if (threadIdx.x < GCN_NCLASS) bs[threadIdx.x] = bl[threadIdx.x];
    __syncthreads();

    int i = blockIdx.x * 256 + threadIdx.x;
    if (i >= n) return;

    float acc[GCN_NCLASS];
    #pragma unroll
    for (int c = 0; c < GCN_NCLASS; ++c) acc[c] = bs[c];

    const float* xr = X + (size_t)i * GCN_D;
    for (int k = 0; k < GCN_D; ++k) {
        float xv = xr[k];
        #pragma unroll
        for (int c = 0; c < GCN_NCLASS; ++c)
            acc[c] = fmaf(xv, wl[k * GCN_NCLASS + c], acc[c]);
    }

    float m = acc[0];
    #pragma unroll
    for (int c = 1; c < GCN_NCLASS; ++c) m = fmaxf(m, acc[c]);
    float sum = 0.0f;
    #pragma unroll
    for (int c = 0; c < GCN_NCLASS; ++c) sum += expf(acc[c] - m);
    float lse = m + logf(sum);
    float* orow = out + (size_t)i * GCN_NCLASS;
    #pragma unroll
    for (int c = 0; c < GCN_NCLASS; ++c) orow[c] = acc[c] - lse;
}

// ---------------------------------------------------------------------------
// Launch: inputs = {x, edge_index, edge_attr, W1, b1, W2, b2, Wl, bl}
// ---------------------------------------------------------------------------
extern "C" void kernel_launch(void* const* d_in, const int* in_sizes, int n_in,
                              void* d_out, int out_size, void* d_ws, size_t ws_size,
                              hipStream_t stream) {
    const float* x   = (const float*)d_in[0];
    const int*   ei  = (const int*)d_in[1];     // [2, E] int32
    const float* W1  = (const float*)d_in[3];
    const float* b1  = (const float*)d_in[4];
    const float* W2  = (const float*)d_in[5];
    const float* b2  = (const float*)d_in[6];
    const float* Wl  = (const float*)d_in[7];
    const float* bl  = (const float*)d_in[8];
    float* out = (float*)d_out;

    const int n = in_sizes[0] / GCN_D;          // 100000
    const int e = in_sizes[1] / 2;              // 1600000
    const int* src = ei;
    const int* dst = ei + e;

    // workspace layout (floats): dinv | bufA (Hs) | bufB (agg / X')
    float* dinv = (float*)d_ws;
    size_t nd_pad = ((size_t)n + 1023) & ~(size_t)1023;
    float* bufA = dinv + nd_pad;
    float* bufB = bufA + (size_t)n * GCN_D;

    const int nfeat = n * GCN_D;
    dim3 blk(256);
    dim3 gN((n + 255) / 256);
    dim3 gE((e + 255) / 256);
    dim3 gF((nfeat + 255) / 256);
    dim3 gEF(((size_t)e * GCN_D + 255) / 256);
    dim3 gGemm((n + 31) / 32);

    // degrees -> dinv (shared by both layers)
    gcn_init_deg<<<gN, blk, 0, stream>>>(dinv, n);
    gcn_deg_accum<<<gE, blk, 0, stream>>>(dst, dinv, e);
    gcn_deg_to_dinv<<<gN, blk, 0, stream>>>(dinv, n);

    // ---- layer 1 ----
    gcn_gemm64_scale<<<gGemm, dim3(64), 0, stream>>>(x, W1, dinv, bufA, n);      // Hs1
    gcn_zero<<<gF, blk, 0, stream>>>(bufB, nfeat);
    gcn_scatter<<<gEF, blk, 0, stream>>>(src, dst, bufA, dinv, bufB, e);
    gcn_finish_conv<<<gF, blk, 0, stream>>>(bufB, bufA, dinv, b1, bufB, n);      // X2

    // ---- layer 2 ----
    gcn_gemm64_scale<<<gGemm, dim3(64), 0, stream>>>(bufB, W2, dinv, bufA, n);   // Hs2
    gcn_zero<<<gF, blk, 0, stream>>>(bufB, nfeat);
    gcn_scatter<<<gEF, blk, 0, stream>>>(src, dst, bufA, dinv, bufB, e);
    gcn_finish_conv<<<gF, blk, 0, stream>>>(bufB, bufA, dinv, b2, bufB, n);      // X3

    // ---- head ----
    gcn_head<<<gN, blk, 0, stream>>>(bufB, Wl, bl, out, n);
}